// LaxCat_7438883357588
// MI455X (gfx1250) — compile-verified
//
#include <hip/hip_runtime.h>
#include <hip/hip_bf16.h>

// ---------------------------------------------------------------------------
// Problem constants (from the reference):
//   B=32, T=512, F=9, H=64, P=C(9,3)=84, K=3, LE=510, LBL=10
// Strategy: collapse the combinatorial conv into a dense K=28 GEMM
// (27 = 9 features x 3 taps, +1 bias row), run all matmuls through
// V_WMMA_F32_16X16X4_F32 (fp32 problem -> fp32 WMMA path), and fold the
// attention-weighted P-reduction into the WMMA accumulator by scaling the
// B-matrix columns per lane. k_hv processes TWO 16-wide l-tiles per wave so
// each Wexp A-fragment load feeds two WMMAs (halves L2 traffic).
// Workspace use ~17.2 MB.
// ---------------------------------------------------------------------------

typedef float v2f __attribute__((ext_vector_type(2)));
typedef float v8f __attribute__((ext_vector_type(8)));

#define NB   32
#define NT   512
#define NF   9
#define NH   64
#define NP   84
#define NLE  510
#define NK   28     // padded reduction dim: 27 data + 1 bias row
#define NPP  96     // P padded to multiple of 16
#define NLBL 10

static __device__ __forceinline__ v8f wmma_f32_k4(v2f a, v2f b, v8f c) {
  // D(16x16 f32) = A(16x4 f32) * B(4x16 f32) + C
  return __builtin_amdgcn_wmma_f32_16x16x4_f32(
      /*neg_a=*/false, a, /*neg_b=*/false, b,
      /*c_mod=*/(short)0, c, /*reuse_a=*/false, /*reuse_b=*/false);
}

// --------------------------------------------------------------------------
// Wexp[p*64+h][28]: dense-embedded conv weights. kappa = f*3+k for the 3
// selected features of combo p; kappa=27 holds conv_b.  grid=84, block=64.
// --------------------------------------------------------------------------
__global__ void k_build_wexp(const float* __restrict__ conv_w,
                             const float* __restrict__ conv_b,
                             float* __restrict__ Wexp) {
  int p = blockIdx.x;
  int h = threadIdx.x;
  // lexicographic combinations(range(9), 3), same order as itertools
  int c0 = 0, c1 = 1, c2 = 2, idx = 0;
  for (int a = 0; a < NF; a++)
    for (int b = a + 1; b < NF; b++)
      for (int c = b + 1; c < NF; c++) {
        if (idx == p) { c0 = a; c1 = b; c2 = c; }
        idx++;
      }
  int fsel[3] = {c0, c1, c2};
  float* row = Wexp + (size_t)(p * NH + h) * NK;
  for (int k = 0; k < NK; k++) row[k] = 0.f;
  for (int i = 0; i < 3; i++)
    for (int k = 0; k < 3; k++)
      row[fsel[i] * 3 + k] = conv_w[(size_t)(p * NH + h) * 9 + i * 3 + k];
  row[27] = conv_b[p * NH + h];
}

// --------------------------------------------------------------------------
// Weff[h][28] = sum_p va_w1[p] * Wexp[p*64+h][.]  (+ va_b1 on the bias row)
// w2ext[64][96] = va_w2 zero-padded in P.          grid=1, block=256.
// --------------------------------------------------------------------------
__global__ void k_build_weff(const float* __restrict__ Wexp,
                             const float* __restrict__ va_w1,
                             const float* __restrict__ va_b1,
                             const float* __restrict__ va_w2,
                             float* __restrict__ Weff,
                             float* __restrict__ w2ext) {
  int t = threadIdx.x;
  if (t < NH) {
    for (int k = 0; k < NK; k++) {
      float s = 0.f;
      for (int p = 0; p < NP; p++)
        s += va_w1[p] * Wexp[(size_t)(p * NH + t) * NK + k];
      if (k == 27) s += va_b1[t];
      Weff[t * NK + k] = s;
    }
  }
  for (int i = t; i < NH * NPP; i += 256) {
    int h = i / NPP, p = i % NPP;
    w2ext[i] = (p < NP) ? va_w2[h * NP + p] : 0.f;
  }
}

// --------------------------------------------------------------------------
// x27[b][kappa][l] = x[b][l+k][f] for kappa=f*3+k (l<510, else 0);
// row 27 = 1.0 for valid l (bias feed).            grid=32, block=512.
// --------------------------------------------------------------------------
__global__ void k_build_x27(const float* __restrict__ x,
                            float* __restrict__ x27) {
  int b = blockIdx.x;
  int l = threadIdx.x;
  float* dst = x27 + (size_t)b * NK * NT;
  bool valid = (l < NLE);
  for (int f = 0; f < NF; f++)
    for (int k = 0; k < 3; k++)
      dst[(size_t)(f * 3 + k) * NT + l] =
          valid ? x[((size_t)b * NT + (l + k)) * NF + f] : 0.f;
  dst[(size_t)27 * NT + l] = valid ? 1.f : 0.f;
}

// --------------------------------------------------------------------------
// t1[b][l][h] = relu( Weff(64x28) @ x27_b(28x16-tile) )  via WMMA.
// grid = B*32 (one 16-wide l-tile per block), block = 32 (one wave).
// --------------------------------------------------------------------------
__global__ void k_t1(const float* __restrict__ x27,
                     const float* __restrict__ Weff,
                     float* __restrict__ t1) {
  int blk = blockIdx.x;
  int b = blk >> 5, lt = blk & 31;
  int lane = threadIdx.x;
  int n = lane & 15, hi = lane >> 4;
  int l0 = lt * 16;
  const float* xb = x27 + (size_t)b * NK * NT;

  v2f bf[7];
#pragma unroll
  for (int f = 0; f < 7; f++) {
    int k0 = 4 * f + (hi ? 2 : 0);
    bf[f].x = xb[(size_t)k0 * NT + l0 + n];
    bf[f].y = xb[(size_t)(k0 + 1) * NT + l0 + n];
  }
#pragma unroll
  for (int mt = 0; mt < 4; mt++) {
    v8f acc = {};
#pragma unroll
    for (int f = 0; f < 7; f++) {
      int k0 = 4 * f + (hi ? 2 : 0);
      v2f a = *(const v2f*)(Weff + (size_t)(mt * 16 + n) * NK + k0);
      acc = wmma_f32_k4(a, bf[f], acc);
    }
#pragma unroll
    for (int r = 0; r < 8; r++) {
      int h = mt * 16 + r + hi * 8;
      t1[((size_t)b * NT + l0 + n) * NH + h] = fmaxf(acc[r], 0.f);
    }
  }
}

// --------------------------------------------------------------------------
// t2 = relu(t1 @ va_w2 + b2); attn = softmax_p(t2), zero-padded to 96.
// grid = B*32 row-tiles of 16, block = 32 (one wave). WMMA over K=64.
// --------------------------------------------------------------------------
__global__ void k_attn(const float* __restrict__ t1,
                       const float* __restrict__ w2ext,
                       const float* __restrict__ va_b2,
                       float* __restrict__ attn) {
  __shared__ float tile[16][NPP];
  int rt = blockIdx.x;
  int lane = threadIdx.x;
  int n = lane & 15, hi = lane >> 4;
  int r0 = rt * 16;

  for (int nt = 0; nt < 6; nt++) {
    v8f acc = {};
    for (int s = 0; s < 16; s++) {
      int k0 = 4 * s + (hi ? 2 : 0);
      v2f a = *(const v2f*)(t1 + (size_t)(r0 + n) * NH + k0);
      v2f bb;
      bb.x = w2ext[(size_t)k0 * NPP + nt * 16 + n];
      bb.y = w2ext[(size_t)(k0 + 1) * NPP + nt * 16 + n];
      acc = wmma_f32_k4(a, bb, acc);
    }
    int col = nt * 16 + n;
    float bias = (col < NP) ? va_b2[col] : 0.f;
#pragma unroll
    for (int r = 0; r < 8; r++)
      tile[r + hi * 8][col] = fmaxf(acc[r] + bias, 0.f);
  }
  __syncthreads();
  if (lane < 16) {
    float m = -1e30f;
    for (int p = 0; p < NP; p++) m = fmaxf(m, tile[lane][p]);
    float s = 0.f;
    for (int p = 0; p < NP; p++) {
      float e = expf(tile[lane][p] - m);
      tile[lane][p] = e;
      s += e;
    }
    float inv = 1.f / s;
    float* dst = attn + (size_t)(r0 + lane) * NPP;
    for (int p = 0; p < NP; p++) dst[p] = tile[lane][p] * inv;
    for (int p = NP; p < NPP; p++) dst[p] = 0.f;
  }
}

// --------------------------------------------------------------------------
// v[b][h][l] = sum_p attn[b,l,p] * conv[b,p,h,l]
//            = sum_p Wexp_p(64x28) @ (attn-col-scaled x27 tile).
// The P-reduction lives in the WMMA accumulator (84 p * 7 k-steps chained).
// TWO 16-wide l-tiles per wave: each Wexp A-fragment load feeds 2 WMMAs.
// grid = B*16, block = 32 (one wave). 4704 WMMAs / 2352 A-loads per block.
// --------------------------------------------------------------------------
__global__ void k_hv(const float* __restrict__ x27,
                     const float* __restrict__ Wexp,
                     const float* __restrict__ attn,
                     float* __restrict__ v) {
  int blk = blockIdx.x;
  int b = blk >> 4, lt = blk & 15;
  int lane = threadIdx.x;
  int n = lane & 15, hi = lane >> 4;
  int l0 = lt * 32;                 // 32 output columns: [l0, l0+32)
  const float* xb = x27 + (size_t)b * NK * NT;

  v2f bf0[7], bf1[7];
#pragma unroll
  for (int f = 0; f < 7; f++) {
    int k0 = 4 * f + (hi ? 2 : 0);
    bf0[f].x = xb[(size_t)k0 * NT + l0 + n];
    bf0[f].y = xb[(size_t)(k0 + 1) * NT + l0 + n];
    bf1[f].x = xb[(size_t)k0 * NT + l0 + 16 + n];
    bf1[f].y = xb[(size_t)(k0 + 1) * NT + l0 + 16 + n];
  }
  const float* arow0 = attn + (size_t)(b * NT + l0 + n) * NPP;
  const float* arow1 = attn + (size_t)(b * NT + l0 + 16 + n) * NPP;

  v8f acc0[4] = {}, acc1[4] = {};
  for (int p = 0; p < NP; p++) {
    float s0 = arow0[p];            // per-column (per-lane) attention weights
    float s1 = arow1[p];
    v2f sb0[7], sb1[7];
#pragma unroll
    for (int f = 0; f < 7; f++) {
      sb0[f] = bf0[f] * s0;
      sb1[f] = bf1[f] * s1;
    }
    const float* Ap = Wexp + (size_t)p * NH * NK;
#pragma unroll
    for (int f = 0; f < 7; f++) {
      int k0 = 4 * f + (hi ? 2 : 0);
#pragma unroll
      for (int mt = 0; mt < 4; mt++) {
        v2f a = *(const v2f*)(Ap + (size_t)(mt * 16 + n) * NK + k0);
        acc0[mt] = wmma_f32_k4(a, sb0[f], acc0[mt]);
        acc1[mt] = wmma_f32_k4(a, sb1[f], acc1[mt]);
      }
    }
  }
#pragma unroll
  for (int mt = 0; mt < 4; mt++) {
#pragma unroll
    for (int r = 0; r < 8; r++) {
      int h = mt * 16 + r + hi * 8;
      v[((size_t)b * NH + h) * NT + l0 + n]      = acc0[mt][r];
      v[((size_t)b * NH + h) * NT + l0 + 16 + n] = acc1[mt][r];
    }
  }
}

// --------------------------------------------------------------------------
// Temporal attention + FC head.  grid = B, block = 64. Negligible cost.
// --------------------------------------------------------------------------
__global__ void k_final(const float* __restrict__ v,
                        const float* __restrict__ ta_w1,
                        const float* __restrict__ ta_b1,
                        const float* __restrict__ ta_w2,
                        const float* __restrict__ ta_b2,
                        const float* __restrict__ fc_w,
                        const float* __restrict__ fc_b,
                        float* __restrict__ out) {
  __shared__ float u1[NH];
  __shared__ float at[NLE];
  __shared__ float red[NH];
  int b = blockIdx.x;
  int t = threadIdx.x;  // 64 threads = h index
  const float* vb = v + (size_t)b * NH * NT;

  float s = 0.f;
  for (int l = 0; l < NLE; l++) s += vb[(size_t)t * NT + l] * ta_w1[l];
  u1[t] = fmaxf(s + ta_b1[t], 0.f);
  __syncthreads();

  for (int l = t; l < NLE; l += 64) {
    float a = 0.f;
    for (int h = 0; h < NH; h++) a += u1[h] * ta_w2[(size_t)h * NLE + l];
    at[l] = fmaxf(a + ta_b2[l], 0.f);
  }
  __syncthreads();

  float m = -1e30f;
  for (int l = t; l < NLE; l += 64) m = fmaxf(m, at[l]);
  red[t] = m;
  __syncthreads();
  if (t == 0) {
    float mm = -1e30f;
    for (int i = 0; i < NH; i++) mm = fmaxf(mm, red[i]);
    red[0] = mm;
  }
  __syncthreads();
  m = red[0];
  __syncthreads();

  float es = 0.f;
  for (int l = t; l < NLE; l += 64) {
    float e = expf(at[l] - m);
    at[l] = e;
    es += e;
  }
  red[t] = es;
  __syncthreads();
  if (t == 0) {
    float ss = 0.f;
    for (int i = 0; i < NH; i++) ss += red[i];
    red[0] = ss;
  }
  __syncthreads();
  float inv = 1.f / red[0];
  __syncthreads();

  float z = 0.f;
  for (int l = 0; l < NLE; l++) z += (at[l] * inv) * vb[(size_t)t * NT + l];
  red[t] = z;
  __syncthreads();
  if (t < NLBL) {
    float o = fc_b[t];
    for (int h = 0; h < NH; h++) o += red[h] * fc_w[t * NH + h];
    out[b * NLBL + t] = o;
  }
}

// --------------------------------------------------------------------------
// Workspace layout (floats). Total = 4,287,232 floats = ~17.2 MB.
// --------------------------------------------------------------------------
static const size_t OFF_WEXP = 0;                       // 84*64*28 = 150528
static const size_t OFF_WEFF = 150528;                  // 64*28    = 1792
static const size_t OFF_W2E  = 152320;                  // 64*96    = 6144
static const size_t OFF_X27  = 158464;                  // 32*28*512= 458752
static const size_t OFF_T1   = 617216;                  // 32*512*64= 1048576
static const size_t OFF_ATT  = 1665792;                 // 32*512*96= 1572864
static const size_t OFF_V    = 3238656;                 // 32*64*512= 1048576

extern "C" void kernel_launch(void* const* d_in, const int* in_sizes, int n_in,
                              void* d_out, int out_size, void* d_ws,
                              size_t ws_size, hipStream_t stream) {
  (void)in_sizes; (void)n_in; (void)out_size; (void)ws_size;
  const float* x      = (const float*)d_in[0];
  const float* conv_w = (const float*)d_in[1];
  const float* conv_b = (const float*)d_in[2];
  const float* va_w1  = (const float*)d_in[3];
  const float* va_b1  = (const float*)d_in[4];
  const float* va_w2  = (const float*)d_in[5];
  const float* va_b2  = (const float*)d_in[6];
  const float* ta_w1  = (const float*)d_in[7];
  const float* ta_b1  = (const float*)d_in[8];
  const float* ta_w2  = (const float*)d_in[9];
  const float* ta_b2  = (const float*)d_in[10];
  const float* fc_w   = (const float*)d_in[11];
  const float* fc_b   = (const float*)d_in[12];
  float* out = (float*)d_out;
  float* ws  = (float*)d_ws;

  float* Wexp  = ws + OFF_WEXP;
  float* Weff  = ws + OFF_WEFF;
  float* w2ext = ws + OFF_W2E;
  float* x27   = ws + OFF_X27;
  float* t1    = ws + OFF_T1;
  float* attn  = ws + OFF_ATT;
  float* v     = ws + OFF_V;

  hipLaunchKernelGGL(k_build_wexp, dim3(NP), dim3(NH), 0, stream,
                     conv_w, conv_b, Wexp);
  hipLaunchKernelGGL(k_build_weff, dim3(1), dim3(256), 0, stream,
                     Wexp, va_w1, va_b1, va_w2, Weff, w2ext);
  hipLaunchKernelGGL(k_build_x27, dim3(NB), dim3(NT), 0, stream, x, x27);
  hipLaunchKernelGGL(k_t1, dim3(NB * 32), dim3(32), 0, stream, x27, Weff, t1);
  hipLaunchKernelGGL(k_attn, dim3(NB * 32), dim3(32), 0, stream,
                     t1, w2ext, va_b2, attn);
  hipLaunchKernelGGL(k_hv, dim3(NB * 16), dim3(32), 0, stream,
                     x27, Wexp, attn, v);
  hipLaunchKernelGGL(k_final, dim3(NB), dim3(64), 0, stream,
                     v, ta_w1, ta_b1, ta_w2, ta_b2, fc_w, fc_b, out);
}